// HashEncoder_69913477644604
// MI455X (gfx1250) — compile-verified
//
#include <hip/hip_runtime.h>
#include <stdint.h>

// ---------------------------------------------------------------------------
// Instant-NGP HashEncoder forward, CDNA5 (gfx1250), wave32.
//   D=3, L=16, C=2, base_res=16, per_level_scale=2, log2_hashmap=19.
//   res[l] = 16<<l (exact). Levels 0..2 dense, 3..15 hashed (mask 0x7FFFF).
//   offsets: 0, 4096, 36864, 299008, then +524288 per level.
// Memory-bound gather kernel: NT hints keep the 56.9MB table resident in the
// 192MB L2; level-0 table (32KB) async-staged to LDS (ASYNCcnt path).
// ---------------------------------------------------------------------------

typedef __attribute__((ext_vector_type(2))) float v2f;
typedef __attribute__((ext_vector_type(4))) float v4f;

#define PRIME_Y 2654435761u
#define PRIME_Z 805459861u
#define HASH_MASK 0x7FFFFu
#define NLVL 16
#define BLOCK 256

struct LvCoord { unsigned ix, iy, iz; float tx, ty, tz; };

__device__ __forceinline__ LvCoord lv_coords(float x, float y, float z, float scale) {
  LvCoord c;
  float px = fmaf(x, scale, 0.5f);
  float py = fmaf(y, scale, 0.5f);
  float pz = fmaf(z, scale, 0.5f);
  float fx = floorf(px), fy = floorf(py), fz = floorf(pz);
  c.tx = px - fx; c.ty = py - fy; c.tz = pz - fz;
  c.ix = (unsigned)fx; c.iy = (unsigned)fy; c.iz = (unsigned)fz;
  return c;
}

__device__ __forceinline__ v2f trilerp(v2f g000, v2f g100, v2f g010, v2f g110,
                                       v2f g001, v2f g101, v2f g011, v2f g111,
                                       float tx, float ty, float tz) {
  float sx = 1.0f - tx, sy = 1.0f - ty, sz = 1.0f - tz;
  v2f a = g000 * (sx * sy * sz);
  a += g100 * (tx * sy * sz);
  a += g010 * (sx * ty * sz);
  a += g110 * (tx * ty * sz);
  a += g001 * (sx * sy * tz);
  a += g101 * (tx * sy * tz);
  a += g011 * (sx * ty * tz);
  a += g111 * (tx * ty * tz);
  return a;
}

// Hashed level: idx = (x ^ y*P2 ^ z*P3) & 0x7FFFF
__device__ __forceinline__ v2f level_hashed(const v2f* __restrict__ eb,
                                            float x, float y, float z, float scale) {
  LvCoord c = lv_coords(x, y, z, scale);
  unsigned hx0 = c.ix,           hx1 = c.ix + 1u;
  unsigned hy0 = c.iy * PRIME_Y, hy1 = hy0 + PRIME_Y;
  unsigned hz0 = c.iz * PRIME_Z, hz1 = hz0 + PRIME_Z;
  v2f g000 = eb[(hx0 ^ hy0 ^ hz0) & HASH_MASK];
  v2f g100 = eb[(hx1 ^ hy0 ^ hz0) & HASH_MASK];
  v2f g010 = eb[(hx0 ^ hy1 ^ hz0) & HASH_MASK];
  v2f g110 = eb[(hx1 ^ hy1 ^ hz0) & HASH_MASK];
  v2f g001 = eb[(hx0 ^ hy0 ^ hz1) & HASH_MASK];
  v2f g101 = eb[(hx1 ^ hy0 ^ hz1) & HASH_MASK];
  v2f g011 = eb[(hx0 ^ hy1 ^ hz1) & HASH_MASK];
  v2f g111 = eb[(hx1 ^ hy1 ^ hz1) & HASH_MASK];
  return trilerp(g000, g100, g010, g110, g001, g101, g011, g111, c.tx, c.ty, c.tz);
}

// Dense level: idx = x + y*res + z*res^2   (works for global or LDS base after
// inlining; InferAddressSpaces promotes the LDS case to ds_load_b64).
template <typename EPtr>
__device__ __forceinline__ v2f level_dense(EPtr eb, float x, float y, float z,
                                           float scale, unsigned r) {
  LvCoord c = lv_coords(x, y, z, scale);
  unsigned rr = r * r;
  unsigned x0 = c.ix,      x1 = c.ix + 1u;
  unsigned y0 = c.iy * r,  y1 = y0 + r;
  unsigned z0 = c.iz * rr, z1 = z0 + rr;
  v2f g000 = eb[x0 + y0 + z0];
  v2f g100 = eb[x1 + y0 + z0];
  v2f g010 = eb[x0 + y1 + z0];
  v2f g110 = eb[x1 + y1 + z0];
  v2f g001 = eb[x0 + y0 + z1];
  v2f g101 = eb[x1 + y0 + z1];
  v2f g011 = eb[x0 + y1 + z1];
  v2f g111 = eb[x1 + y1 + z1];
  return trilerp(g000, g100, g010, g110, g001, g101, g011, g111, c.tx, c.ty, c.tz);
}

__global__ __launch_bounds__(BLOCK)
void hashenc_fwd_kernel(const float* __restrict__ inp,
                        const float* __restrict__ emb,
                        float* __restrict__ out, int B) {
  // ---- stage level-0 dense table (4096 x float2 = 32 KB) into LDS ----------
  __shared__ float s_l0[8192];
  {
    const unsigned lbase = (unsigned)(uintptr_t)(void*)s_l0;   // low32 of generic = LDS addr
    const unsigned long long gbase = (unsigned long long)(uintptr_t)emb;
#pragma unroll
    for (int i = 0; i < 8; ++i) {
      unsigned byteoff = ((unsigned)threadIdx.x + (unsigned)i * BLOCK) * 16u; // b128 chunks
      unsigned ldsa = lbase + byteoff;
      unsigned long long ga = gbase + (unsigned long long)byteoff;
      // CDNA5 async global->LDS copy (ASYNCcnt path), GV addressing.
      asm volatile("global_load_async_to_lds_b128 %0, %1, off"
                   :: "v"(ldsa), "v"(ga) : "memory");
    }
    asm volatile("s_wait_asynccnt 0" ::: "memory");
  }
  __syncthreads();

  const int t = blockIdx.x * BLOCK + threadIdx.x;
  if (t >= B) return;

  // ---- streaming input (non-temporal: don't pollute L2) --------------------
  const float* ip = inp + 3 * (size_t)t;
  float x = (__builtin_nontemporal_load(ip + 0) + 1.0f) * 0.5f;
  float y = (__builtin_nontemporal_load(ip + 1) + 1.0f) * 0.5f;
  float z = (__builtin_nontemporal_load(ip + 2) + 1.0f) * 0.5f;

  const v2f* e = (const v2f*)emb;
  v2f acc[NLVL];

  // Level 0 (res=16, dense, LDS-resident)
  acc[0] = level_dense(((const v2f*)s_l0), x, y, z, 15.0f, 16u);
  // Level 1 (res=32, dense, offset 4096)
  acc[1] = level_dense(e + 4096u, x, y, z, 31.0f, 32u);
  // Level 2 (res=64, dense, offset 36864)
  acc[2] = level_dense(e + 36864u, x, y, z, 63.0f, 64u);

  // Levels 3..15: hashed, table size 2^19 each, offsets 299008 + (l-3)*524288
#pragma unroll
  for (int l = 3; l < NLVL; ++l) {
    const float scale = (float)((16u << l) - 1u);
    const unsigned off = 299008u + (unsigned)(l - 3) * 524288u;
    acc[l] = level_hashed(e + off, x, y, z, scale);
  }

  // ---- streaming output [B,32], 8 x b128 non-temporal stores per point -----
  v4f* o4 = (v4f*)out + (size_t)t * 8;
#pragma unroll
  for (int k = 0; k < 8; ++k) {
    v4f v;
    v.x = acc[2 * k].x;     v.y = acc[2 * k].y;
    v.z = acc[2 * k + 1].x; v.w = acc[2 * k + 1].y;
    __builtin_nontemporal_store(v, o4 + k);
  }
}

extern "C" void kernel_launch(void* const* d_in, const int* in_sizes, int n_in,
                              void* d_out, int out_size, void* d_ws, size_t ws_size,
                              hipStream_t stream) {
  const float* inp = (const float*)d_in[0];   // [B, 3] float32
  const float* emb = (const float*)d_in[1];   // [7114752, 2] float32
  float* out = (float*)d_out;                 // [B, 32] float32
  const int B = in_sizes[0] / 3;
  const int grid = (B + BLOCK - 1) / BLOCK;
  hipLaunchKernelGGL(hashenc_fwd_kernel, dim3(grid), dim3(BLOCK), 0, stream,
                     inp, emb, out, B);
}